// MCPA_33706903339586
// MI455X (gfx1250) — compile-verified
//
#include <hip/hip_runtime.h>
#include <hip/hip_bf16.h>

// Problem constants (reference shapes are fixed).
#define N_   16
#define C_   9
#define NC_  (N_ * C_)          // 144
#define HW_  (512 * 512)        // 262144
#define TOPN 0.15f
#define CAP_ 32768              // candidate cap per (n,c); >=4x analytic worst case

// ---------------------------------------------------------------------------
// Helpers
// ---------------------------------------------------------------------------

// Monotonic float -> uint key: unsigned compare order == float compare order.
__device__ __forceinline__ unsigned fkey(float v) {
    unsigned u = __float_as_uint(v);
    return (u & 0x80000000u) ? ~u : (u | 0x80000000u);
}
__device__ __forceinline__ float key_to_float(unsigned key) {
    unsigned u = (key & 0x80000000u) ? (key ^ 0x80000000u) : ~key;
    return __uint_as_float(u);
}

// LDS byte offset of a generic pointer that points into LDS.
typedef __attribute__((address_space(3))) char lds_char;
__device__ __forceinline__ unsigned lds_off_of(const void* p) {
    return (unsigned)(size_t)(lds_char*)p;
}

// CDNA5 async copy: 16B/lane global -> LDS (GVS addressing).
__device__ __forceinline__ void async_copy16(unsigned ldsOff, unsigned byteOff,
                                             const float* base) {
    asm volatile("global_load_async_to_lds_b128 %0, %1, %2"
                 :
                 : "v"(ldsOff), "v"(byteOff), "s"(base)
                 : "memory");
}
__device__ __forceinline__ void wait_async_le1() {
    asm volatile("s_wait_asynccnt 1" ::: "memory");
}
__device__ __forceinline__ void wait_async_0() {
    asm volatile("s_wait_asynccnt 0" ::: "memory");
}

__device__ __forceinline__ int compute_k(float r) {
    float fp = floorf(r * (float)HW_);   // f_p = floor(ratio*H*W)
    return (int)floorf(fp * TOPN);       // k = floor(f_p * 0.15)
}

// Descending-order rank selection over an LDS histogram. All 256 threads
// participate; result (bin, remaining rank) is broadcast via shared outs.
// part must point to a shared int[256]. Requires 1 <= target <= sum(h).
template <int NB>
__device__ __forceinline__ void block_select(const int* h, int* part, int target,
                                             int tid, int* outBin, int* outRank) {
    constexpr int CS = NB / 256;
    const int topBin = NB - 1 - tid * CS;
    int sum = 0;
    #pragma unroll
    for (int j = 0; j < CS; ++j) sum += h[topBin - j];
    part[tid] = sum;
    __syncthreads();
    for (int off = 1; off < 256; off <<= 1) {      // inclusive scan, top-down
        int t = (tid >= off) ? part[tid - off] : 0;
        __syncthreads();
        if (tid >= off) part[tid] += t;
        __syncthreads();
    }
    const int incl = part[tid];
    const int excl = incl - sum;
    if (excl < target && incl >= target) {         // exactly one thread
        int acc = excl;
        for (int j = 0; j < CS; ++j) {
            const int b = topBin - j;
            acc += h[b];
            if (acc >= target) {
                *outBin  = b;
                *outRank = target - (acc - h[b]);
                break;
            }
        }
    }
    __syncthreads();   // outs visible to all
}

// ---------------------------------------------------------------------------
// Histogram pass (radix level 12 / 12 / 8 bits). grid=(32,144), block=256.
// Input staged via double-buffered global_load_async_to_lds_b128.
// ---------------------------------------------------------------------------
template <int LEVEL>
__global__ void hist_kernel(const float* __restrict__ inp,
                            const unsigned* __restrict__ selKey,
                            int* __restrict__ histOut) {
    constexpr int NB = (LEVEL == 2) ? 256 : 4096;

    __shared__ __align__(16) float stage[2][1024];
    __shared__ int hist[NB];

    const int tid = threadIdx.x;
    const int nc  = blockIdx.y;

    for (int b = tid; b < NB; b += 256) hist[b] = 0;
    __syncthreads();

    unsigned sel = 0;
    if (LEVEL > 0) sel = selKey[nc];

    const float* base = inp + (size_t)nc * HW_ + (size_t)blockIdx.x * 8192;
    const unsigned lds0 = lds_off_of(&stage[0][tid * 4]);
    const unsigned lds1 = lds_off_of(&stage[1][tid * 4]);
    const unsigned gOff = (unsigned)tid * 16u;

    async_copy16(lds0, gOff, base);
    #pragma unroll 1
    for (int it = 0; it < 8; ++it) {
        const int buf = it & 1;
        if (it + 1 < 8) {
            async_copy16(buf ? lds0 : lds1, gOff + (unsigned)(it + 1) * 4096u, base);
            wait_async_le1();
        } else {
            wait_async_0();
        }
        float4 v = *reinterpret_cast<const float4*>(&stage[buf][tid * 4]);
        float vals[4] = {v.x, v.y, v.z, v.w};
        #pragma unroll
        for (int j = 0; j < 4; ++j) {
            unsigned key = fkey(vals[j]);
            if (LEVEL == 0) {
                atomicAdd(&hist[key >> 20], 1);
            } else if (LEVEL == 1) {
                if ((key >> 20) == sel) atomicAdd(&hist[(key >> 8) & 0xFFFu], 1);
            } else {
                if ((key >> 8) == sel) atomicAdd(&hist[key & 0xFFu], 1);
            }
        }
    }
    __syncthreads();

    for (int b = tid; b < NB; b += 256) {
        int c = hist[b];
        if (c) atomicAdd(&histOut[(size_t)nc * NB + b], c);
    }
}

// ---------------------------------------------------------------------------
// Selection kernels for the histogram path. grid=144, block=256.
// ---------------------------------------------------------------------------
template <int LEVEL>
__global__ void select_kernel(const int* __restrict__ hist,
                              const float* __restrict__ ratio,
                              unsigned* __restrict__ selKey,
                              int* __restrict__ rankArr,
                              float* __restrict__ thrArr) {
    constexpr int NB = (LEVEL == 2) ? 256 : 4096;
    __shared__ int part[256];
    __shared__ int sBin, sRank;

    const int nc  = blockIdx.x;
    const int n   = nc / C_;
    const int tid = threadIdx.x;

    const int k = compute_k(ratio[n]);
    if (k == 0) {
        if (LEVEL == 0 && tid == 0) {
            thrArr[nc] = 1.0f;
            selKey[nc] = 0xFFFFFFFFu;   // sentinel: matches no real key prefix
        }
        return;
    }

    const int target = (LEVEL == 0) ? k : rankArr[nc];
    const int* h = hist + (size_t)nc * NB;

    block_select<NB>(h, part, target, tid, &sBin, &sRank);

    if (tid == 0) {
        if (LEVEL == 0) {
            selKey[nc]  = (unsigned)sBin;
            rankArr[nc] = sRank;
        } else if (LEVEL == 1) {
            selKey[nc]  = (selKey[nc] << 12) | (unsigned)sBin;
            rankArr[nc] = sRank;
        } else {
            thrArr[nc] = key_to_float((selKey[nc] << 8) | (unsigned)sBin);
        }
    }
}

// ---------------------------------------------------------------------------
// Compaction pass (fast path): stream the plane, collect elements whose
// top-12-bit key matches selKey into cand[nc][CAP_]. Matches are gathered in
// an LDS list; one global atomicAdd per block reserves the output range.
// ---------------------------------------------------------------------------
__global__ void compact_kernel(const float* __restrict__ inp,
                               const unsigned* __restrict__ selKey,
                               float* __restrict__ cand,
                               int* __restrict__ candCount) {
    __shared__ __align__(16) float stage[2][1024];
    __shared__ float list[8192];   // block covers exactly 8192 elements
    __shared__ int lcnt, gbase;

    const int tid = threadIdx.x;
    const int nc  = blockIdx.y;
    if (tid == 0) lcnt = 0;
    __syncthreads();

    const unsigned sel = selKey[nc];
    const float* base = inp + (size_t)nc * HW_ + (size_t)blockIdx.x * 8192;
    const unsigned lds0 = lds_off_of(&stage[0][tid * 4]);
    const unsigned lds1 = lds_off_of(&stage[1][tid * 4]);
    const unsigned gOff = (unsigned)tid * 16u;

    async_copy16(lds0, gOff, base);
    #pragma unroll 1
    for (int it = 0; it < 8; ++it) {
        const int buf = it & 1;
        if (it + 1 < 8) {
            async_copy16(buf ? lds0 : lds1, gOff + (unsigned)(it + 1) * 4096u, base);
            wait_async_le1();
        } else {
            wait_async_0();
        }
        float4 v = *reinterpret_cast<const float4*>(&stage[buf][tid * 4]);
        float vals[4] = {v.x, v.y, v.z, v.w};
        #pragma unroll
        for (int j = 0; j < 4; ++j) {
            if ((fkey(vals[j]) >> 20) == sel) {
                int p = atomicAdd(&lcnt, 1);
                list[p] = vals[j];
            }
        }
    }
    __syncthreads();

    if (tid == 0) gbase = lcnt ? atomicAdd(&candCount[nc], lcnt) : 0;
    __syncthreads();

    const int cnt = lcnt, cb = gbase;
    for (int i = tid; i < cnt; i += 256) {
        const int pos = cb + i;
        if (pos < CAP_) cand[(size_t)nc * CAP_ + pos] = list[i];   // clamped: no OOB
    }
}

// ---------------------------------------------------------------------------
// Finalize (fast path): resolve remaining 20 key bits from the candidate
// list. Two LDS-histogram selection rounds; one block per (n,c).
// ---------------------------------------------------------------------------
__global__ void finalize_kernel(const float* __restrict__ cand,
                                const int* __restrict__ candCount,
                                const float* __restrict__ ratio,
                                const unsigned* __restrict__ selKey,
                                const int* __restrict__ rankArr,
                                float* __restrict__ thrArr) {
    __shared__ int hist[4096];
    __shared__ int part[256];
    __shared__ int sBin, sRank;

    const int nc  = blockIdx.x;
    const int n   = nc / C_;
    const int tid = threadIdx.x;

    const int k = compute_k(ratio[n]);
    if (k == 0) return;   // thr already = 1.0 from select<0>

    int cnt = candCount[nc];
    if (cnt > CAP_) cnt = CAP_;
    const int target = rankArr[nc];
    const float* cl = cand + (size_t)nc * CAP_;

    // Round B: key bits [19:8]
    for (int b = tid; b < 4096; b += 256) hist[b] = 0;
    __syncthreads();
    for (int i = tid; i < cnt; i += 256) {
        unsigned key = fkey(cl[i]);
        atomicAdd(&hist[(key >> 8) & 0xFFFu], 1);
    }
    __syncthreads();
    block_select<4096>(hist, part, target, tid, &sBin, &sRank);
    const unsigned b2 = (unsigned)sBin;
    const int r2 = sRank;
    __syncthreads();

    // Round C: key bits [7:0], filtered by b2
    for (int b = tid; b < 256; b += 256) hist[b] = 0;
    __syncthreads();
    for (int i = tid; i < cnt; i += 256) {
        unsigned key = fkey(cl[i]);
        if (((key >> 8) & 0xFFFu) == b2) atomicAdd(&hist[key & 0xFFu], 1);
    }
    __syncthreads();
    block_select<256>(hist, part, r2, tid, &sBin, &sRank);

    if (tid == 0) {
        const unsigned key = (selKey[nc] << 20) | (b2 << 8) | (unsigned)sBin;
        thrArr[nc] = key_to_float(key);
    }
}

// ---------------------------------------------------------------------------
// Mask pass: out[n,p] = (any_c inp[n,c,p] > thr[n,c]) ? 0 : x[n,p]
// grid=(256,16), block=256; one float4 per thread.
// ---------------------------------------------------------------------------
__global__ void mask_kernel(const float* __restrict__ inp,
                            const float* __restrict__ x,
                            const float* __restrict__ thrArr,
                            float* __restrict__ out) {
    __shared__ float thr[C_];
    const int n = blockIdx.y;
    if (threadIdx.x < C_) thr[threadIdx.x] = thrArr[n * C_ + threadIdx.x];
    __syncthreads();

    const int p4 = blockIdx.x * 256 + threadIdx.x;
    const float4 xv = reinterpret_cast<const float4*>(x + (size_t)n * HW_)[p4];

    int m0 = 0, m1 = 0, m2 = 0, m3 = 0;
    #pragma unroll
    for (int c = 0; c < C_; ++c) {
        const float4 v =
            reinterpret_cast<const float4*>(inp + (size_t)(n * C_ + c) * HW_)[p4];
        const float t = thr[c];
        m0 |= (v.x > t);
        m1 |= (v.y > t);
        m2 |= (v.z > t);
        m3 |= (v.w > t);
    }
    float4 o;
    o.x = m0 ? 0.0f : xv.x;
    o.y = m1 ? 0.0f : xv.y;
    o.z = m2 ? 0.0f : xv.z;
    o.w = m3 ? 0.0f : xv.w;
    reinterpret_cast<float4*>(out + (size_t)n * HW_)[p4] = o;
}

// ---------------------------------------------------------------------------
// Launcher: pick compaction fast path if scratch allows, else 3-pass path.
// ---------------------------------------------------------------------------
extern "C" void kernel_launch(void* const* d_in, const int* in_sizes, int n_in,
                              void* d_out, int out_size, void* d_ws, size_t ws_size,
                              hipStream_t stream) {
    (void)in_sizes; (void)n_in; (void)out_size;

    const float* inp   = (const float*)d_in[0];   // [16,9,512,512]
    const float* xim   = (const float*)d_in[1];   // [16,1,512,512]
    const float* ratio = (const float*)d_in[2];   // [16]
    float* out = (float*)d_out;                   // [16,1,512,512]

    // Scratch layout (4-byte units)
    int*      hist1     = (int*)d_ws;                     // NC*4096
    int*      hist2     = hist1 + (size_t)NC_ * 4096;     // NC*4096
    int*      hist3     = hist2 + (size_t)NC_ * 4096;     // NC*256
    unsigned* selKey    = (unsigned*)(hist3 + (size_t)NC_ * 256);
    int*      rankArr   = (int*)(selKey + NC_);
    float*    thrArr    = (float*)(rankArr + NC_);
    int*      candCount = (int*)(thrArr + NC_);
    float*    cand      = (float*)(candCount + NC_);

    const size_t stateInts  = (size_t)NC_ * 4096 * 2 + (size_t)NC_ * 256 + (size_t)NC_ * 4;
    const size_t fastBytes  = (stateInts + (size_t)NC_ * CAP_) * 4;

    hipMemsetAsync(d_ws, 0, stateInts * 4, stream);

    const dim3 hgrid(32, NC_);

    // Pass 1 + selection of the 12-bit bin (both paths).
    hist_kernel<0><<<hgrid, 256, 0, stream>>>(inp, selKey, hist1);
    select_kernel<0><<<NC_, 256, 0, stream>>>(hist1, ratio, selKey, rankArr, thrArr);

    if (ws_size >= fastBytes) {
        // Fast path: compact candidates, finalize from the compact list.
        compact_kernel<<<hgrid, 256, 0, stream>>>(inp, selKey, cand, candCount);
        finalize_kernel<<<NC_, 256, 0, stream>>>(cand, candCount, ratio,
                                                 selKey, rankArr, thrArr);
    } else {
        // Fallback: two more full histogram passes.
        hist_kernel<1><<<hgrid, 256, 0, stream>>>(inp, selKey, hist2);
        select_kernel<1><<<NC_, 256, 0, stream>>>(hist2, ratio, selKey, rankArr, thrArr);
        hist_kernel<2><<<hgrid, 256, 0, stream>>>(inp, selKey, hist3);
        select_kernel<2><<<NC_, 256, 0, stream>>>(hist3, ratio, selKey, rankArr, thrArr);
    }

    mask_kernel<<<dim3(256, N_), 256, 0, stream>>>(inp, xim, thrArr, out);
}